// BallModel_91242285236773
// MI455X (gfx1250) — compile-verified
//
#include <hip/hip_runtime.h>

// pos(t) = p0 + (v0*dt)*t + (g*dt^2)*t(t-1)/2 ; traj[0] = p0 (exact at t=0).
//
// WMMA rank-3 factorization with t = 8m + s, M = 8m:
//   (M+s)(M+s-1)/2 = M(M-1)/2 + s(s-1)/2 + M*s
//   pos_c(8m + s) = [p + v's + a's(s-1)/2] * 1
//                 + [v' + a's]             * (8m)
//                 + [a']                   * (8m)(8m-1)/2
// A(16x4): row m = [1, 8m, 8m(8m-1)/2, 0]   (constant across tiles, exact fp32)
// B(4x16): col n=(j,c), s = tile*128 + j    (fp64 -> fp32, recomputed per tile)
// D(16x16) = one V_WMMA_F32_16X16X4_F32 => 128 timesteps x {x,y} = 1KB out.
//
// Each wave32 processes TILES_PER_WAVE tiles to amortize fixed overhead
// (input loads, fp64 constants, address setup) over 8 WMMAs / 8KB of stores.

typedef __attribute__((ext_vector_type(2))) float v2f;
typedef __attribute__((ext_vector_type(8))) float v8f;

#define TILES_PER_WAVE 8

__global__ __launch_bounds__(256) void ball_traj_wmma(
    const float* __restrict__ pos0,   // [2]
    const float* __restrict__ vel0,   // [2]
    float* __restrict__ out,          // [num_tiles*256] floats
    int num_tiles)
{
    const int wave  = blockIdx.x * (blockDim.x >> 5) + (threadIdx.x >> 5);
    const int tile0 = wave * TILES_PER_WAVE;
    if (tile0 >= num_tiles) return;            // wave-uniform

    const int lane  = threadIdx.x & 31;
    const int half  = lane >> 4;               // K-half for A/B, M-half for D
    const int idx16 = lane & 15;               // M row (A), N col (B/D)

    // ---- A matrix (32-bit A 16x4 layout: lanes 0-15 K=0,1; lanes 16-31 K=2,3)
    const float mf = (float)idx16;
    v2f a;
    if (half == 0) { a.x = 1.0f;                            a.y = 8.0f * mf; }
    else           { a.x = 4.0f * mf * (8.0f * mf - 1.0f);  a.y = 0.0f;      }

    // ---- tile-invariant fp64 coefficients (per lane: component c = idx16&1)
    const int    j  = idx16 >> 1;              // s-offset within tile
    const int    c  = idx16 & 1;               // component: 0=x, 1=y
    const double dt = 0.01;
    const double g  = (c == 0) ? 0.0 : -9.81;
    const double p  = (double)pos0[c];
    const double vd = (double)vel0[c] * dt;    // v' = v0*dt
    const double ad = g * dt * dt;             // a' = g*dt^2
    const float  adf = (float)ad;

    float* outw = out + (size_t)tile0 * 256 + (size_t)half * 128 + (size_t)idx16;
    const double s_base = (double)tile0 * 128.0 + (double)j;

#pragma unroll
    for (int tt = 0; tt < TILES_PER_WAVE; ++tt) {
        if (tile0 + tt >= num_tiles) break;    // wave-uniform; EXEC stays all-1s

        // ---- B matrix column n=idx16 (lanes 0-15: K=0,1; lanes 16-31: K=2,3)
        const double s   = s_base + 128.0 * (double)tt;
        const double tri = s * (s - 1.0) * 0.5;
        const float  b0f = (float)(p + vd * s + ad * tri);   // B[0][n]
        const float  b1f = (float)(vd + ad * s);             // B[1][n]
        v2f b;
        b.x = half ? adf  : b0f;               // B[2][n] : B[0][n]
        b.y = half ? 0.0f : b1f;               // B[3][n] : B[1][n]

        // ---- D = A x B + 0  (v_wmma_f32_16x16x4_f32)
        v8f acc = {};
        acc = __builtin_amdgcn_wmma_f32_16x16x4_f32(
            /*neg_a=*/false, a, /*neg_b=*/false, b,
            /*c_mod=*/(short)0, acc, /*reuse_a=*/false, /*reuse_b=*/false);

        // D lane/VGPR (m = r + 8*half, n = idx16) -> dword offset
        //   16*r + 128*half + idx16 in the tile's 256-dword (1KB) block:
        // each store_b32 writes two full contiguous 64B cachelines.
        float* outp = outw + (size_t)tt * 256;
#pragma unroll
        for (int r = 0; r < 8; ++r)
            outp[16 * r] = acc[r];
    }
}

// Scalar fallback for any elements beyond the last full 256-float tile
// (not hit for SIM_STEPS = 10,000,000, which is exactly 78,125 tiles).
__global__ void ball_traj_tail(const float* __restrict__ pos0,
                               const float* __restrict__ vel0,
                               float* __restrict__ out,
                               int start_elem, int total_elems)
{
    int i = start_elem + blockIdx.x * blockDim.x + threadIdx.x;
    if (i >= total_elems) return;
    int t = i >> 1, c = i & 1;
    double dt = 0.01;
    double g  = c ? -9.81 : 0.0;
    double td = (double)t;
    double v  = (double)pos0[c] + (double)vel0[c] * dt * td
              + g * dt * dt * (td * (td - 1.0) * 0.5);
    out[i] = (float)v;
}

extern "C" void kernel_launch(void* const* d_in, const int* in_sizes, int n_in,
                              void* d_out, int out_size, void* d_ws, size_t ws_size,
                              hipStream_t stream)
{
    // setup_inputs order: [0]=ball_mass (unused), [1]=initial_position[2], [2]=initial_velocity[2]
    const float* pos0 = (const float*)d_in[1];
    const float* vel0 = (const float*)d_in[2];
    float* out = (float*)d_out;

    const int tiles = out_size / 256;                       // 78,125 for 10M steps
    if (tiles > 0) {
        const int waves  = (tiles + TILES_PER_WAVE - 1) / TILES_PER_WAVE;
        const int waves_per_block = 8;                      // 256 threads = 8 wave32
        const int blocks = (waves + waves_per_block - 1) / waves_per_block;
        ball_traj_wmma<<<blocks, 256, 0, stream>>>(pos0, vel0, out, tiles);
    }
    const int done = tiles * 256;
    if (done < out_size) {
        const int rem = out_size - done;
        ball_traj_tail<<<(rem + 255) / 256, 256, 0, stream>>>(pos0, vel0, out, done, out_size);
    }
}